// ModelSGLang_68186900792113
// MI455X (gfx1250) — compile-verified
//
#include <hip/hip_runtime.h>

// Gather:  rows = kv_buffer[loc]; out_nope = rows[:, :512]; out_rope = rows[:, 512:576]
// kv_buffer: (131072, 1, 576) f32, loc: (65536,) int32, outputs concatenated flat:
//   d_out[0 .. 65536*512)            = nope
//   d_out[65536*512 .. 65536*576)    = rope
// Pure data movement: ~302 MB HBM traffic -> ~13 us floor at 23.3 TB/s. No FLOPs -> no WMMA;
// the CDNA5 path that matters is the ASYNCcnt-tracked global<->LDS DMA (B128) + NT store hints.

#define N_ROWS    65536
#define ROW_F     576
#define NOPE_F    512
#define NOPE_CH   (NOPE_F / 4)                 // 128 float4 chunks / row
#define ROPE_CH   ((ROW_F - NOPE_F) / 4)       // 16  float4 chunks / row
#define NOPE_TOT  (N_ROWS * NOPE_CH)           // 8,388,608 chunks
#define ROPE_TOT  (N_ROWS * ROPE_CH)           // 1,048,576 chunks
#define BLK       256                          // 8 wave32 per block
#define CPT       8                            // chunks per thread (4 KB async batch per wave)
#define NBLK_NOPE (NOPE_TOT / (BLK * CPT))     // 4096 blocks, exact
#define NBLK_ROPE (ROPE_TOT / (BLK * CPT))     // 512 blocks, exact

typedef float v4f __attribute__((ext_vector_type(4)));
typedef int   v4i __attribute__((ext_vector_type(4)));
typedef __attribute__((address_space(3))) v4i* lds_v4i_p;

#if defined(__HIP_DEVICE_COMPILE__)
#if __has_builtin(__builtin_amdgcn_global_load_async_to_lds_b128) && \
    __has_builtin(__builtin_amdgcn_global_store_async_from_lds_b128) && \
    __has_builtin(__builtin_amdgcn_s_wait_asynccnt)
#define USE_ASYNC 1
#else
#define USE_ASYNC 0
#endif
#endif

// ---------------- nope region: chunks j in [0, NOPE_TOT) ----------------
// 128 chunks/row, stride NBLK_NOPE*BLK = 1,048,576 (multiple of 128): every wave32's 32
// consecutive j share one row per k -> readfirstlane legally scalarizes loc[] to s_load.
__global__ __launch_bounds__(BLK) void gather_nope_kernel(const float* __restrict__ kv,
                                                          const int*   __restrict__ loc,
                                                          float*       __restrict__ out)
{
#if defined(__HIP_DEVICE_COMPILE__)
  const int stride = NBLK_NOPE * BLK;
  const int j0 = (int)blockIdx.x * BLK + (int)threadIdx.x;

  const float* srcs[CPT];
  float*       dsts[CPT];
#pragma unroll
  for (int k = 0; k < CPT; ++k) {                       // all loc fetches issue back-to-back
    int j   = j0 + k * stride;
    int row = __builtin_amdgcn_readfirstlane(j >> 7);   // wave-uniform -> s_load
    long long base = (long long)loc[row] * ROW_F;
    srcs[k] = kv + base + ((j & (NOPE_CH - 1)) << 2);
    dsts[k] = out + ((long long)j << 2);
  }

#if USE_ASYNC
  __shared__ v4f stage[BLK * CPT];                      // 32 KB of 320 KB/WGP
#pragma unroll
  for (int k = 0; k < CPT; ++k)
    __builtin_amdgcn_global_load_async_to_lds_b128(
        (v4i*)srcs[k], (lds_v4i_p)&stage[(int)threadIdx.x + k * BLK], 0, /*RT*/0);
  __builtin_amdgcn_s_wait_asynccnt(0);
#pragma unroll
  for (int k = 0; k < CPT; ++k)
    __builtin_amdgcn_global_store_async_from_lds_b128(
        (v4i*)dsts[k], (lds_v4i_p)&stage[(int)threadIdx.x + k * BLK], 0, /*TH_NT*/1);
  // S_ENDPGM waits idle -> async stores drain before retire.
#else
#pragma unroll
  for (int k = 0; k < CPT; ++k) {
    v4f v = *(const v4f*)srcs[k];
    __builtin_nontemporal_store(v, (v4f*)dsts[k]);
  }
#endif
#endif
}

// ---------------- rope region: chunks j in [0, ROPE_TOT) ----------------
// 16 chunks/row -> 2 rows per wave per k; keep vector loc load (L2-resident, 0.25 MB).
__global__ __launch_bounds__(BLK) void gather_rope_kernel(const float* __restrict__ kv,
                                                          const int*   __restrict__ loc,
                                                          float*       __restrict__ out)
{
#if defined(__HIP_DEVICE_COMPILE__)
  const int stride = NBLK_ROPE * BLK;
  const int j0 = (int)blockIdx.x * BLK + (int)threadIdx.x;
  float* const out_rope = out + (long long)N_ROWS * NOPE_F;

  const float* srcs[CPT];
  float*       dsts[CPT];
#pragma unroll
  for (int k = 0; k < CPT; ++k) {
    int j   = j0 + k * stride;
    int row = j >> 4;
    long long base = (long long)loc[row] * ROW_F + NOPE_F;
    srcs[k] = kv + base + ((j & (ROPE_CH - 1)) << 2);
    dsts[k] = out_rope + ((long long)j << 2);
  }

#if USE_ASYNC
  __shared__ v4f stage[BLK * CPT];
#pragma unroll
  for (int k = 0; k < CPT; ++k)
    __builtin_amdgcn_global_load_async_to_lds_b128(
        (v4i*)srcs[k], (lds_v4i_p)&stage[(int)threadIdx.x + k * BLK], 0, /*RT*/0);
  __builtin_amdgcn_s_wait_asynccnt(0);
#pragma unroll
  for (int k = 0; k < CPT; ++k)
    __builtin_amdgcn_global_store_async_from_lds_b128(
        (v4i*)dsts[k], (lds_v4i_p)&stage[(int)threadIdx.x + k * BLK], 0, /*TH_NT*/1);
#else
#pragma unroll
  for (int k = 0; k < CPT; ++k) {
    v4f v = *(const v4f*)srcs[k];
    __builtin_nontemporal_store(v, (v4f*)dsts[k]);
  }
#endif
#endif
}

extern "C" void kernel_launch(void* const* d_in, const int* in_sizes, int n_in,
                              void* d_out, int out_size, void* d_ws, size_t ws_size,
                              hipStream_t stream) {
  (void)in_sizes; (void)n_in; (void)out_size; (void)d_ws; (void)ws_size;
  const float* kv  = (const float*)d_in[0];
  const int*   loc = (const int*)d_in[1];
  float*       out = (float*)d_out;
  gather_nope_kernel<<<NBLK_NOPE, BLK, 0, stream>>>(kv, loc, out);
  gather_rope_kernel<<<NBLK_ROPE, BLK, 0, stream>>>(kv, loc, out);
}